// LESForceStressOutput_31379031065338
// MI455X (gfx1250) — compile-verified
//
#include <hip/hip_runtime.h>
#include <hip/hip_bf16.h>

typedef __attribute__((ext_vector_type(2))) float v2f;
typedef __attribute__((ext_vector_type(8))) float v8f;

#define N_GRAPHS_MAX 64
#define GV_FLOATS (N_GRAPHS_MAX * 6)

// Kernel 1: force = -pos_grad (written straight into d_out) and zero the
// per-graph virial accumulator in workspace (must re-zero every call).
__global__ void les_init_kernel(const float* __restrict__ pos_grad,
                                float* __restrict__ force,
                                float* __restrict__ gvir, int n3) {
  int i = blockIdx.x * blockDim.x + threadIdx.x;
  if (i < n3) force[i] = -pos_grad[i];
  if (i < GV_FLOATS) gvir[i] = 0.f;
}

// Kernel 2: edge pass. 38.4M global f32 atomics for force; virial goes through
// a per-block LDS table (ds_add_f32) and is flushed with 384 global atomics
// per block at the end (persistent grid keeps flush traffic negligible).
__global__ void les_edge_kernel(const float* __restrict__ rij,
                                const float* __restrict__ fij,
                                const int* __restrict__ eidx,
                                const int* __restrict__ batch,
                                float* __restrict__ force,
                                float* __restrict__ gvir, int E) {
  __shared__ float lvir[GV_FLOATS];
  for (int i = threadIdx.x; i < GV_FLOATS; i += blockDim.x) lvir[i] = 0.f;
  __syncthreads();

  int stride = gridDim.x * blockDim.x;
  for (int e = blockIdx.x * blockDim.x + threadIdx.x; e < E; e += stride) {
    float rx = rij[3 * e + 0], ry = rij[3 * e + 1], rz = rij[3 * e + 2];
    float fx = fij[3 * e + 0], fy = fij[3 * e + 1], fz = fij[3 * e + 2];
    int i = eidx[e];        // edge_idx[0][e]
    int j = eidx[E + e];    // edge_idx[1][e]
    // force += fij scattered to i, -= fij scattered to j
    unsafeAtomicAdd(&force[3 * i + 0], fx);
    unsafeAtomicAdd(&force[3 * i + 1], fy);
    unsafeAtomicAdd(&force[3 * i + 2], fz);
    unsafeAtomicAdd(&force[3 * j + 0], -fx);
    unsafeAtomicAdd(&force[3 * j + 1], -fy);
    unsafeAtomicAdd(&force[3 * j + 2], -fz);
    // fused node->graph virial: graph of receiving node j
    int g = batch[j];
    float* p = &lvir[g * 6];
    atomicAdd(p + 0, rx * fx);  // xx
    atomicAdd(p + 1, ry * fy);  // yy
    atomicAdd(p + 2, rz * fz);  // zz
    atomicAdd(p + 3, rx * fy);  // xy
    atomicAdd(p + 4, ry * fz);  // yz
    atomicAdd(p + 5, rz * fx);  // zx
  }
  __syncthreads();
  for (int i = threadIdx.x; i < GV_FLOATS; i += blockDim.x)
    unsafeAtomicAdd(&gvir[i], lvir[i]);
}

// Kernel 3: per-graph stress. One wave per graph; lr_raw = les_cell^T @ cell_grad
// done with V_WMMA_F32_16X16X4_F32 (3x3 padded into 16x4 x 4x16, exact fp32).
// Output: stress = -(virial + lr_raw_voigt) / V.
__global__ void les_stress_kernel(const float* __restrict__ cell_grad,
                                  const float* __restrict__ les_cell,
                                  const float* __restrict__ vol,
                                  const float* __restrict__ gvir,
                                  float* __restrict__ out) {
  int g = blockIdx.x;
  int lane = threadIdx.x;           // 0..31, full wave32
  const float* C = les_cell + g * 9;   // row-major 3x3
  const float* G = cell_grad + g * 9;  // row-major 3x3

  // ISA A-layout 16x4 f32: lane m (0-15) holds K=0,1; lane m+16 holds K=2,3.
  // ISA B-layout 4x16:     VGPR0 = row K=0 (lanes 0-15) / K=2 (lanes 16-31),
  //                        VGPR1 = row K=1 / K=3.
  int m = lane & 15;                // A row / B col
  int hi = lane >> 4;
  int k0 = hi * 2, k1 = k0 + 1;     // K indices for the two VGPRs
  int mc = (m < 2) ? m : 2;         // clamped (mask handles validity)
  int k0c = (k0 < 2) ? k0 : 2;
  int k1c = (k1 < 2) ? k1 : 2;
  float rowm = (m < 3) ? 1.f : 0.f;
  float m0 = (k0 < 3) ? rowm : 0.f;
  float m1 = (k1 < 3) ? rowm : 0.f;

  v2f a, b;
  a.x = m0 * C[k0c * 3 + mc];       // A[m][k] = (C^T)[m][k] = C[k][m]
  a.y = m1 * C[k1c * 3 + mc];
  b.x = m0 * G[k0c * 3 + mc];       // B[k][n] = G[k][n], n == m here
  b.y = m1 * G[k1c * 3 + mc];
  v8f c = {0.f, 0.f, 0.f, 0.f, 0.f, 0.f, 0.f, 0.f};
  v8f d = __builtin_amdgcn_wmma_f32_16x16x4_f32(
      /*neg_a=*/false, a, /*neg_b=*/false, b,
      /*c_mod=*/(short)0, c, /*reuse_a=*/false, /*reuse_b=*/false);

  // D layout: VGPR r = row r, lanes 0-15 are cols 0-15. Pull 3x3 corner via LDS.
  __shared__ float lr[9];
  if (lane < 3) {
    lr[0 * 3 + lane] = d[0];
    lr[1 * 3 + lane] = d[1];
    lr[2 * 3 + lane] = d[2];
  }
  __syncthreads();
  if (lane == 0) {
    float inv = -1.0f / vol[g];
    const float* v = gvir + g * 6;
    float* o = out + g * 6;
    o[0] = inv * (v[0] + lr[0]);  // xx
    o[1] = inv * (v[1] + lr[4]);  // yy
    o[2] = inv * (v[2] + lr[8]);  // zz
    o[3] = inv * (v[3] + lr[1]);  // xy
    o[4] = inv * (v[4] + lr[5]);  // yz
    o[5] = inv * (v[5] + lr[2]);  // zx (lr[0][2])
  }
}

extern "C" void kernel_launch(void* const* d_in, const int* in_sizes, int n_in,
                              void* d_out, int out_size, void* d_ws, size_t ws_size,
                              hipStream_t stream) {
  const float* rij      = (const float*)d_in[0];
  const float* fij      = (const float*)d_in[1];
  const float* pos_grad = (const float*)d_in[2];
  const float* cell_grad = (const float*)d_in[3];
  const float* les_cell  = (const float*)d_in[4];
  const float* vol       = (const float*)d_in[5];
  const int*   eidx      = (const int*)d_in[6];
  const int*   batch     = (const int*)d_in[7];

  int E  = in_sizes[0] / 3;   // 6.4M edges
  int n3 = in_sizes[2];       // N*3 = 600000
  int G  = in_sizes[5];       // 64 graphs

  float* force  = (float*)d_out;       // [N*3]
  float* stress = force + n3;          // [G*6]
  float* gvir   = (float*)d_ws;        // [G*6] scratch accumulators

  const int threads = 256;
  les_init_kernel<<<(n3 + threads - 1) / threads, threads, 0, stream>>>(
      pos_grad, force, gvir, n3);
  les_edge_kernel<<<2048, threads, 0, stream>>>(
      rij, fij, eidx, batch, force, gvir, E);
  les_stress_kernel<<<G, 32, 0, stream>>>(
      cell_grad, les_cell, vol, gvir, stress);
}